// singleGATLayer_13426067767844
// MI455X (gfx1250) — compile-verified
//
#include <hip/hip_runtime.h>
#include <hip/hip_bf16.h>

typedef __attribute__((ext_vector_type(2))) float v2f;
typedef __attribute__((ext_vector_type(8))) float v8f;

#define F 128           // IN_F == OUT_F == 128
#define SLOPE 0.2f
#define WAVES_PER_BLOCK 4

// ---------------------------------------------------------------------------
// Kernel 0: zero h' accumulator (d_out) and rowsum
// ---------------------------------------------------------------------------
__global__ void gat_zero_kernel(float* __restrict__ hp, float* __restrict__ rowsum,
                                int nNodes) {
    int gid = blockIdx.x * blockDim.x + threadIdx.x;
    if (gid < nNodes * F) hp[gid] = 0.0f;
    if (gid < nNodes)     rowsum[gid] = 0.0f;
}

// ---------------------------------------------------------------------------
// Kernel 1: h = X @ W^T via V_WMMA_F32_16X16X4_F32, fused s1 = h@a1, s2 = h@a2
//
// FP32 WMMA fragment layouts (ISA 7.12.2):
//   A (16x4):  lane L -> M = L%16;  VGPR v -> K = 2*(L/16) + v
//              => per lane: two consecutive floats of X row M at col k0+2*(L/16)
//   B (4x16):  lane L -> N = L%16;  VGPR v -> K = 2*(L/16) + v
//              B[k][n] = W[n][k]  => two consecutive floats of W row N
//   C/D (16x16): lane L, VGPR r -> M = r + 8*(L/16), N = L%16
//
// One wave computes a 16-row tile of h across all 128 output features
// (8 accumulators), reusing each A fragment 8x. K loop: 128/4 = 32 steps.
// ---------------------------------------------------------------------------
__global__ void gat_gemm_kernel(const float* __restrict__ X,   // [nNodes,128]
                                const float* __restrict__ W,   // [128,128]
                                const float* __restrict__ a,   // [256]
                                float* __restrict__ h,         // [nNodes,128]
                                float* __restrict__ s1,        // [nNodes]
                                float* __restrict__ s2,        // [nNodes]
                                int nTiles) {
    const int lane  = threadIdx.x & 31;
    const int wave  = threadIdx.x >> 5;
    const int mtile = blockIdx.x * WAVES_PER_BLOCK + wave;
    if (mtile >= nTiles) return;                 // wave-uniform: EXEC all-1 inside

    const int half = lane >> 4;                  // 0: lanes 0-15, 1: lanes 16-31
    const int l16  = lane & 15;

    const float* xrow = X + (size_t)(mtile * 16 + l16) * F;

    // attention vector coefficients for this lane's column set
    float a1c[8], a2c[8];
#pragma unroll
    for (int nt = 0; nt < 8; ++nt) {
        a1c[nt] = a[nt * 16 + l16];
        a2c[nt] = a[F + nt * 16 + l16];
    }

    v8f acc[8];
#pragma unroll
    for (int nt = 0; nt < 8; ++nt) acc[nt] = (v8f){0,0,0,0,0,0,0,0};

    for (int k0 = 0; k0 < F; k0 += 4) {
        __builtin_prefetch(xrow + k0 + 16, 0, 0);            // global_prefetch_b8
        v2f av = *(const v2f*)(xrow + k0 + 2 * half);        // A fragment
#pragma unroll
        for (int nt = 0; nt < 8; ++nt) {
            const float* wrow = W + (size_t)(nt * 16 + l16) * F;
            v2f bv = *(const v2f*)(wrow + k0 + 2 * half);    // B fragment
            acc[nt] = __builtin_amdgcn_wmma_f32_16x16x4_f32(
                false, av, false, bv, (short)0, acc[nt], false, false);
        }
    }

    // epilogue: store h tile, fused dot with a1/a2 + 16-lane tree reduction
#pragma unroll
    for (int r = 0; r < 8; ++r) {
        const int m = mtile * 16 + 8 * half + r;
        float p1 = 0.0f, p2 = 0.0f;
#pragma unroll
        for (int nt = 0; nt < 8; ++nt) {
            float v = acc[nt][r];
            h[(size_t)m * F + nt * 16 + l16] = v;
            p1 += v * a1c[nt];
            p2 += v * a2c[nt];
        }
#pragma unroll
        for (int msk = 1; msk <= 8; msk <<= 1) {             // reduce within 16-lane group
            p1 += __shfl_xor(p1, msk, 32);
            p2 += __shfl_xor(p2, msk, 32);
        }
        if (l16 == 0) { s1[m] = p1; s2[m] = p2; }
    }
}

// ---------------------------------------------------------------------------
// Kernel 2: per-edge score -> alpha, accumulate rowsum[src]
// ---------------------------------------------------------------------------
__global__ void gat_edge_kernel(const int* __restrict__ ei,   // [nEdges,2]
                                const float* __restrict__ s1,
                                const float* __restrict__ s2,
                                float* __restrict__ alpha,
                                float* __restrict__ rowsum,
                                int nEdges) {
    int e = blockIdx.x * blockDim.x + threadIdx.x;
    if (e >= nEdges) return;
    int src = ei[2 * e + 0];
    int dst = ei[2 * e + 1];
    float sc = s1[src] + s2[dst];
    float lr = sc > 0.0f ? sc : SLOPE * sc;      // leaky_relu
    float al = __expf(-lr);
    al = fminf(al, 5.0f);                        // clip(-5,5): al > 0 always
    alpha[e] = al;
    unsafeAtomicAdd(&rowsum[src], al);           // global_atomic_add_f32
}

// ---------------------------------------------------------------------------
// Kernel 3: scatter  h'[src] += alpha[e] * h[dst]   (one thread per edge-feature)
// h (25.6 MB) is fully L2-resident (192 MB L2); atomics resolve at L2.
// ---------------------------------------------------------------------------
__global__ void gat_scatter_kernel(const int* __restrict__ ei,
                                   const float* __restrict__ alpha,
                                   const float* __restrict__ h,
                                   float* __restrict__ hp,
                                   int nEdges) {
    long long gid = (long long)blockIdx.x * blockDim.x + threadIdx.x;
    int e = (int)(gid >> 7);
    int f = (int)(gid & (F - 1));
    if (e >= nEdges) return;
    int src = ei[2 * e + 0];
    int dst = ei[2 * e + 1];
    float v = alpha[e] * h[(size_t)dst * F + f];
    unsafeAtomicAdd(&hp[(size_t)src * F + f], v);
}

// ---------------------------------------------------------------------------
// Kernel 4: h' = elu(h' / max(rowsum, 1e-8))   in place on d_out
// ---------------------------------------------------------------------------
__global__ void gat_final_kernel(float* __restrict__ hp,
                                 const float* __restrict__ rowsum,
                                 int total) {
    int gid = blockIdx.x * blockDim.x + threadIdx.x;
    if (gid >= total) return;
    int node = gid >> 7;
    float rs = rowsum[node];
    rs = rs > 0.0f ? rs : 1e-8f;
    float v = hp[gid] / rs;
    hp[gid] = v > 0.0f ? v : expm1f(v);
}

// ---------------------------------------------------------------------------
// Launch
//   d_in[0] input [nNodes,128] f32 | d_in[1] W [128,128] f32
//   d_in[2] a [256,1] f32          | d_in[3] edge_index [nEdges,2] i32
//   d_out: [nNodes,128] f32 (doubles as h' accumulator)
//   d_ws layout (floats): h | s1 | s2 | rowsum | alpha   (~29.5 MB)
// ---------------------------------------------------------------------------
extern "C" void kernel_launch(void* const* d_in, const int* in_sizes, int n_in,
                              void* d_out, int out_size, void* d_ws, size_t ws_size,
                              hipStream_t stream) {
    const float* X  = (const float*)d_in[0];
    const float* W  = (const float*)d_in[1];
    const float* a  = (const float*)d_in[2];
    const int*   ei = (const int*)d_in[3];

    const int nNodes = in_sizes[0] / F;
    const int nEdges = in_sizes[3] / 2;

    float* hp = (float*)d_out;
    float* ws = (float*)d_ws;
    float* h      = ws;
    float* s1     = h  + (size_t)nNodes * F;
    float* s2     = s1 + nNodes;
    float* rowsum = s2 + nNodes;
    float* alpha  = rowsum + nNodes;

    const int total = nNodes * F;

    gat_zero_kernel<<<(total + 255) / 256, 256, 0, stream>>>(hp, rowsum, nNodes);

    const int nTiles = (nNodes + 15) / 16;       // 50000/16 = 3125 exactly
    const int gBlocks = (nTiles + WAVES_PER_BLOCK - 1) / WAVES_PER_BLOCK;
    gat_gemm_kernel<<<gBlocks, 32 * WAVES_PER_BLOCK, 0, stream>>>(
        X, W, a, h, s1, s2, nTiles);

    gat_edge_kernel<<<(nEdges + 255) / 256, 256, 0, stream>>>(
        ei, s1, s2, alpha, rowsum, nEdges);

    const long long scThreads = (long long)nEdges * F;
    gat_scatter_kernel<<<(int)((scThreads + 255) / 256), 256, 0, stream>>>(
        ei, alpha, h, hp, nEdges);

    gat_final_kernel<<<(total + 255) / 256, 256, 0, stream>>>(hp, rowsum, total);
}